// GAT_89215060673063
// MI455X (gfx1250) — compile-verified
//
#include <hip/hip_runtime.h>
#include <math.h>

typedef __attribute__((ext_vector_type(2))) float v2f;
typedef __attribute__((ext_vector_type(8))) float v8f;

#define NEG_SLOPE 0.2f

__device__ __forceinline__ void atomicMaxF(float* addr, float val) {
  // IEEE-754 ordering trick; valid for all sign combinations.
  if (val >= 0.0f) (void)atomicMax((int*)addr, __float_as_int(val));
  else             (void)atomicMin((unsigned int*)addr, __float_as_uint(val));
}

// ---------------------------------------------------------------------------
// C[M,Ncol] = A[M,K] x B[K,Ncol], fp32, via V_WMMA_F32_16X16X4_F32.
// Requirements: K <= 128, K % 8 == 0, Ncol <= 128, Ncol % 16 == 0, M % 16 == 0.
// 256 threads = 8 wave32s; one 16x16 C tile per wave.
//
// B is staged in LDS in a K-paired layout: pair index p = (k>>1)*Ncol + n
// holds {B[k][n], B[k+1][n]} contiguously, so each lane's B fragment for a
// WMMA k-step is a single aligned ds_load_b64. A fragments are 8-byte
// global loads. Two independent accumulator chains break the WMMA RAW chain.
// ---------------------------------------------------------------------------
__global__ __launch_bounds__(256) void gemm_f32_wmma(
    const float* __restrict__ A, const float* __restrict__ B,
    float* __restrict__ C, int M, int K, int Ncol)
{
  __shared__ float sB[128 * 128];  // 64 KB, paired layout (v2f per pair index)
  const int tid = threadIdx.x;

  // Cooperative staging: thread handles pair index p = k2*Ncol + n,
  // loads B[2*k2][n] and B[2*k2+1][n], stores one contiguous b64.
  {
    v2f* s2 = (v2f*)sB;
    const int nPairs = (K >> 1) * Ncol;
    for (int p = tid; p < nPairs; p += 256) {
      int k2 = p / Ncol;
      int n  = p - k2 * Ncol;
      v2f t;
      t.x = B[(2 * k2) * Ncol + n];
      t.y = B[(2 * k2 + 1) * Ncol + n];
      s2[p] = t;
    }
  }
  __syncthreads();

  const int wave   = tid >> 5;
  const int lane   = tid & 31;
  const int laneHi = lane >> 4;   // 0: K = k0..k0+1, 1: K = k0+2..k0+3
  const int l16    = lane & 15;

  const int colTiles = Ncol >> 4;
  const int rowTiles = M >> 4;
  const int tile = blockIdx.x * 8 + wave;
  if (tile >= rowTiles * colTiles) return;   // wave-uniform branch
  const int rowBase = (tile / colTiles) << 4;
  const int colBase = (tile % colTiles) << 4;

  const float* Arow = A + (size_t)(rowBase + l16) * K + 2 * laneHi;
  const v2f*   Bp   = ((const v2f*)sB) + (colBase + l16) + (size_t)laneHi * Ncol;

  v8f acc0 = {}, acc1 = {};
  for (int k0 = 0; k0 < K; k0 += 8) {
    // chain 0: K-step at k0
    v2f a0 = *(const v2f*)(Arow + k0);
    v2f b0 = Bp[(size_t)(k0 >> 1) * Ncol];
    acc0 = __builtin_amdgcn_wmma_f32_16x16x4_f32(false, a0, false, b0,
                                                 (short)0, acc0, false, false);
    // chain 1: K-step at k0+4 (independent accumulator)
    v2f a1 = *(const v2f*)(Arow + k0 + 4);
    v2f b1 = Bp[(size_t)((k0 >> 1) + 2) * Ncol];
    acc1 = __builtin_amdgcn_wmma_f32_16x16x4_f32(false, a1, false, b1,
                                                 (short)0, acc1, false, false);
  }
  v8f acc = acc0 + acc1;

  // C/D layout: VGPR r -> row (rowBase + r + 8*laneHi), col (colBase + l16)
  float* Crow = C + (size_t)(rowBase + 8 * laneHi) * Ncol + colBase + l16;
#pragma unroll
  for (int r = 0; r < 8; ++r) Crow[(size_t)r * Ncol] = acc[r];
}

// ---------------------------------------------------------------------------
// Elementwise / edge kernels
// ---------------------------------------------------------------------------
__global__ void fill_f32(float* __restrict__ p, float v, int n) {
  int i = blockIdx.x * blockDim.x + threadIdx.x;
  if (i < n) p[i] = v;
}

// el[n,h] = dot(feat[n,h,:], al[h,:]); er likewise. feat is [N,H,d] contiguous.
__global__ void attn_scores(const float* __restrict__ feat,
                            const float* __restrict__ al,
                            const float* __restrict__ ar,
                            float* __restrict__ el, float* __restrict__ er,
                            int NH, int H, int d)
{
  int i = blockIdx.x * blockDim.x + threadIdx.x;
  if (i >= NH) return;
  int h = i % H;
  const float* f = feat + (size_t)i * d;
  float sl = 0.f, sr = 0.f;
  for (int j = 0; j < d; ++j) {
    float v = f[j];
    sl += v * al[h * d + j];
    sr += v * ar[h * d + j];
  }
  el[i] = sl;
  er[i] = sr;
}

// e[i,h] = leaky_relu(el[src,h] + er[dst,h]); running max into m[dst,h].
__global__ void edge_logits(const int* __restrict__ src, const int* __restrict__ dst,
                            const float* __restrict__ el, const float* __restrict__ er,
                            float* __restrict__ e, float* __restrict__ m,
                            int E, int H)
{
  int i = blockIdx.x * blockDim.x + threadIdx.x;
  if (i >= E * H) return;
  int ed = i / H, h = i - ed * H;
  int s = src[ed], t = dst[ed];
  float v = el[s * H + h] + er[t * H + h];
  v = (v > 0.f) ? v : NEG_SLOPE * v;
  e[i] = v;
  atomicMaxF(&m[t * H + h], v);
}

// e[i] <- exp(e[i] - m[dst]); denominators accumulated per dst.
__global__ void edge_exp(const int* __restrict__ dst, float* __restrict__ e,
                         const float* __restrict__ m, float* __restrict__ den,
                         int E, int H)
{
  int i = blockIdx.x * blockDim.x + threadIdx.x;
  if (i >= E * H) return;
  int ed = i / H, h = i - ed * H;
  int t = dst[ed];
  float ex = __expf(e[i] - m[t * H + h]);
  e[i] = ex;
  atomicAdd(&den[t * H + h], ex);
}

// agg[dst, c] += feat[src, c] * (ex[edge,h] / den[dst,h]),  c = h*d + j
__global__ void edge_aggregate(const int* __restrict__ src, const int* __restrict__ dst,
                               const float* __restrict__ feat,
                               const float* __restrict__ ex,
                               const float* __restrict__ den,
                               float* __restrict__ agg, int E, int H, int d)
{
  const int C = H * d;
  long long i = (long long)blockIdx.x * blockDim.x + threadIdx.x;
  if (i >= (long long)E * C) return;
  int ed = (int)(i / C);
  int c  = (int)(i - (long long)ed * C);
  int h  = c / d;
  int s = src[ed], t = dst[ed];
  float alpha = ex[(size_t)ed * H + h] / den[(size_t)t * H + h];
  atomicAdd(&agg[(size_t)t * C + c], feat[(size_t)s * C + c] * alpha);
}

__global__ void bias_elu(const float* __restrict__ agg, const float* __restrict__ b,
                         float* __restrict__ out, int total, int C)
{
  int i = blockIdx.x * blockDim.x + threadIdx.x;
  if (i >= total) return;
  float x = agg[i] + b[i % C];
  out[i] = (x > 0.f) ? x : (__expf(x) - 1.0f);
}

__global__ void bias_add(float* __restrict__ out, const float* __restrict__ b,
                         int total, int C)
{
  int i = blockIdx.x * blockDim.x + threadIdx.x;
  if (i >= total) return;
  out[i] += b[i % C];
}

// ---------------------------------------------------------------------------
// Host launcher
// ---------------------------------------------------------------------------
static inline unsigned cdiv(long long a, long long b) { return (unsigned)((a + b - 1) / b); }

extern "C" void kernel_launch(void* const* d_in, const int* in_sizes, int n_in,
                              void* d_out, int out_size, void* d_ws, size_t ws_size,
                              hipStream_t stream)
{
  const float* features = (const float*)d_in[0];
  const float* W1  = (const float*)d_in[1];
  const float* a1l = (const float*)d_in[2];
  const float* a1r = (const float*)d_in[3];
  const float* b1  = (const float*)d_in[4];
  const float* W2  = (const float*)d_in[5];
  const float* a2l = (const float*)d_in[6];
  const float* a2r = (const float*)d_in[7];
  const float* b2  = (const float*)d_in[8];
  const int*   src = (const int*)d_in[9];
  const int*   dst = (const int*)d_in[10];

  const int Fin = 128, H1 = 4, hid = 32, C1 = H1 * hid; // 128
  const int Fout = 16;
  const int N = in_sizes[0] / Fin;
  const int E = in_sizes[9];

  // Workspace carve-out (256B aligned).
  char* ws = (char*)d_ws;
  size_t off = 0;
  auto alloc = [&](size_t bytes) -> float* {
    float* p = (float*)(ws + off);
    off += (bytes + 255) & ~(size_t)255;
    return p;
  };
  float* feat1 = alloc((size_t)N * C1 * 4);
  float* el1   = alloc((size_t)N * H1 * 4);
  float* er1   = alloc((size_t)N * H1 * 4);
  float* e1    = alloc((size_t)E * H1 * 4);
  float* m1    = alloc((size_t)N * H1 * 4);
  float* den1  = alloc((size_t)N * H1 * 4);
  float* agg1  = alloc((size_t)N * C1 * 4);
  // Layer-2 aliases (layer-1 buffers are dead by the time these are written):
  float* h1    = feat1;  // elu(agg1+b1); feat1 dead after aggregation
  float* feat2 = agg1;   // h1 @ W2; agg1 dead after bias_elu
  float* el2   = el1;
  float* er2   = er1;
  float* e2    = e1;
  float* m2    = m1;
  float* den2  = den1;
  float* out   = (float*)d_out;

  const int B = 256;

  // ---- Layer 1 ----
  // feat1 = features @ W1   [N,128]
  {
    unsigned tiles = (unsigned)((N / 16) * (C1 / 16));
    gemm_f32_wmma<<<cdiv(tiles, 8), 256, 0, stream>>>(features, W1, feat1, N, Fin, C1);
  }
  attn_scores<<<cdiv((long long)N * H1, B), B, 0, stream>>>(feat1, a1l, a1r, el1, er1,
                                                            N * H1, H1, hid);
  fill_f32<<<cdiv((long long)N * H1, B), B, 0, stream>>>(m1, -INFINITY, N * H1);
  fill_f32<<<cdiv((long long)N * H1, B), B, 0, stream>>>(den1, 0.f, N * H1);
  fill_f32<<<cdiv((long long)N * C1, B), B, 0, stream>>>(agg1, 0.f, N * C1);
  edge_logits<<<cdiv((long long)E * H1, B), B, 0, stream>>>(src, dst, el1, er1, e1, m1, E, H1);
  edge_exp<<<cdiv((long long)E * H1, B), B, 0, stream>>>(dst, e1, m1, den1, E, H1);
  edge_aggregate<<<cdiv((long long)E * C1, B), B, 0, stream>>>(src, dst, feat1, e1, den1,
                                                               agg1, E, H1, hid);
  bias_elu<<<cdiv((long long)N * C1, B), B, 0, stream>>>(agg1, b1, h1, N * C1, C1);

  // ---- Layer 2 ----
  // feat2 = h1 @ W2   [N,16]
  {
    unsigned tiles = (unsigned)((N / 16) * (Fout / 16));
    gemm_f32_wmma<<<cdiv(tiles, 8), 256, 0, stream>>>(h1, W2, feat2, N, C1, Fout);
  }
  attn_scores<<<cdiv((long long)N, B), B, 0, stream>>>(feat2, a2l, a2r, el2, er2, N, 1, Fout);
  fill_f32<<<cdiv((long long)N, B), B, 0, stream>>>(m2, -INFINITY, N);
  fill_f32<<<cdiv((long long)N, B), B, 0, stream>>>(den2, 0.f, N);
  fill_f32<<<cdiv((long long)N * Fout, B), B, 0, stream>>>(out, 0.f, N * Fout);
  edge_logits<<<cdiv((long long)E, B), B, 0, stream>>>(src, dst, el2, er2, e2, m2, E, 1);
  edge_exp<<<cdiv((long long)E, B), B, 0, stream>>>(dst, e2, m2, den2, E, 1);
  edge_aggregate<<<cdiv((long long)E * Fout, B), B, 0, stream>>>(src, dst, feat2, e2, den2,
                                                                 out, E, 1, Fout);
  bias_add<<<cdiv((long long)N * Fout, B), B, 0, stream>>>(out, b2, N * Fout, Fout);
}